// Q2FPhys_backbone_23407571763540
// MI455X (gfx1250) — compile-verified
//
#include <hip/hip_runtime.h>

// ============================================================================
// Q2FPhys backbone for MI455X (gfx1250, wave32, WMMA).
//  - All convs = implicit GEMM on V_WMMA_F32_16X16X4_F32 (fp32 matrix pipe).
//  - Weights packed [tap][outCh][16 inCh] so K-chunks are tap-uniform and
//    channel-adjacent: A/B fragment loads are single b64 loads, no div/mod in
//    the inner loops (the previous version burned ~40 VALU ops per WMMA on
//    magic-number division).
//  - Temporal-difference channels computed on the fly via a per-block LDS
//    channel table (no 250MB diffs tensor).
//  - BN+bias+ReLU folded into per-channel scale/bias epilogues.
//  - Mamba projections via generic WMMA GEMM (C = A*W^T [+C]).
//  - Selective scan: block per (n,d), 48 states in lanes, LDS reduce per step.
//  - seq lives in d_out; stages update in place.
// ============================================================================

#define EPS_ 1e-5f

typedef float v2f __attribute__((ext_vector_type(2)));
typedef float v8f __attribute__((ext_vector_type(8)));

__device__ __forceinline__ v8f wmma4(v2f a, v2f b, v8f c) {
  // V_WMMA_F32_16X16X4_F32 : D = A(16x4) * B(4x16) + C(16x16), all f32
  return __builtin_amdgcn_wmma_f32_16x16x4_f32(false, a, false, b, (short)0, c,
                                               false, false);
}

__device__ __forceinline__ float sigm_(float x) { return 1.f / (1.f + __expf(-x)); }
__device__ __forceinline__ v2f ldv2g(const float* p) { return *(const v2f*)p; }

// ---------------------------------------------------------------------------
// Prep kernels: weight packing [tap][o][16c] and BN folding
// ---------------------------------------------------------------------------
__global__ void k_pack_w1(const float* __restrict__ w11, const float* __restrict__ w12,
                          float* __restrict__ out) {
  int i = blockIdx.x * blockDim.x + threadIdx.x;
  if (i >= 49 * 16 * 16) return;
  int c = i & 15, o = (i >> 4) & 15, t = i >> 8;
  float v = 0.f;
  if (o < 8 && c < 3) v = w11[(o * 3 + c) * 49 + t];
  else if (o >= 8 && c >= 3 && c < 15) v = w12[((o - 8) * 12 + (c - 3)) * 49 + t];
  out[i] = v;
}

__global__ void k_pack_w2(const float* __restrict__ w21, const float* __restrict__ w22,
                          float* __restrict__ out) {
  int i = blockIdx.x * blockDim.x + threadIdx.x;
  if (i >= 49 * 32 * 16) return;
  int c = i & 15, o = (i >> 4) & 31, t = i >> 9;
  float v = 0.f;
  if (o < 16 && c < 8) v = w21[(o * 8 + c) * 49 + t];
  else if (o >= 16 && c >= 8) v = w22[((o - 16) * 8 + (c - 8)) * 49 + t];
  out[i] = v;
}

__global__ void k_pack_w3(const float* __restrict__ w3, float* __restrict__ out) {
  int i = blockIdx.x * blockDim.x + threadIdx.x;
  if (i >= 75 * 64 * 16) return;
  int c = i & 15, o = (i >> 4) & 63, t = i >> 10;
  out[i] = w3[o * 1200 + c * 75 + t];  // (64,16,3,5,5); t = kd*25+kh*5+kw
}

__global__ void k_make_sb(const float* __restrict__ g, const float* __restrict__ b,
                          const float* __restrict__ m, const float* __restrict__ v,
                          const float* __restrict__ convb,
                          float* __restrict__ sOut, float* __restrict__ bOut, int n) {
  int i = blockIdx.x * blockDim.x + threadIdx.x;
  if (i >= n) return;
  float sc = g[i] * rsqrtf(v[i] + EPS_);
  sOut[i] = sc;
  bOut[i] = (convb[i] - m[i]) * sc + b[i];
}

__global__ void k_negexp(const float* __restrict__ in, float* __restrict__ out, int n) {
  int i = blockIdx.x * blockDim.x + threadIdx.x;
  if (i < n) out[i] = -__expf(in[i]);
}

// ---------------------------------------------------------------------------
// Stem conv1: fused conv11(3ch)+conv12(12 diff-ch), 7x7 stride2 pad3, BN+ReLU.
// Virtual input = 16 channels {x(3), diffs(12), zero(1)} built via LDS table.
// x: (2,3,160,128,128). out: (320, 64*64, 16) NHWC.  K = 49 taps * 16 ch.
// ---------------------------------------------------------------------------
__global__ void k_conv1(const float* __restrict__ x, const float* __restrict__ Wp,
                        const float* __restrict__ sb, float* __restrict__ out) {
  extern __shared__ float lw[];            // 49*16*16 packed weights
  __shared__ size_t tA[16], tB[16];        // per-channel base offsets into x
  __shared__ float tsA[16], tsB[16];       // per-channel scales (0/1)
  for (int i = threadIdx.x; i < 49 * 16 * 16; i += blockDim.x) lw[i] = Wp[i];

  int b = blockIdx.y, n = b / 160, d = b - (b / 160) * 160;
  if (threadIdx.x < 16) {
    int c = threadIdx.x;
    int d1 = (d - 1 > 0) ? d - 1 : 0;
    int d2 = (d - 2 > 0) ? d - 2 : 0;
    int d4 = (d + 1 < 159) ? d + 1 : 159;
    int d5 = (d + 2 < 159) ? d + 2 : 159;
    int fa[4] = {d1, d, d4, d5};
    int fb[4] = {d2, d1, d, d4};
    int cc = 0, fA = 0, fB = 0;
    float sA = 0.f, sB = 0.f;
    if (c < 3) { cc = c; fA = d; fB = d; sA = 1.f; sB = 0.f; }
    else if (c < 15) {
      int dc = c - 3, g = dc / 3;
      cc = dc - g * 3; fA = fa[g]; fB = fb[g]; sA = 1.f; sB = 1.f;
    }
    tA[c] = ((size_t)(n * 3 + cc) * 160 + fA) * 16384;
    tB[c] = ((size_t)(n * 3 + cc) * 160 + fB) * 16384;
    tsA[c] = sA; tsB[c] = sB;
  }
  __syncthreads();

  int wid = threadIdx.x >> 5, lane = threadIdx.x & 31;
  int tile = blockIdx.x * 4 + wid;
  int base = tile * 16;
  int m = lane & 15;
  int p = base + m, oh = p >> 6, ow = p & 63;
  int khalf = (lane < 16) ? 0 : 2;
  v8f acc = {};
#pragma unroll 1
  for (int kh = 0; kh < 7; ++kh) {
    int ih = oh * 2 + kh - 3;
#pragma unroll 1
    for (int kw = 0; kw < 7; ++kw) {
      int iw = ow * 2 + kw - 3;
      bool inb = (ih >= 0 && ih < 128 && iw >= 0 && iw < 128);
      size_t sp = inb ? ((size_t)ih * 128 + iw) : 0;
      float msk = inb ? 1.f : 0.f;
      const float* lwt = lw + ((kh * 7 + kw) * 16 + m) * 16;
#pragma unroll
      for (int cb = 0; cb < 16; cb += 4) {
        int cbase = cb + khalf;
        v2f a, bb;
#pragma unroll
        for (int j = 0; j < 2; ++j) {
          int c = cbase + j;
          a[j] = msk * (tsA[c] * x[tA[c] + sp] - tsB[c] * x[tB[c] + sp]);
        }
        bb = *(const v2f*)&lwt[cbase];
        acc = wmma4(a, bb, acc);
      }
    }
  }
  float sc = sb[m], bi = sb[16 + m];
  int rbase = (lane < 16) ? 0 : 8;
#pragma unroll
  for (int r = 0; r < 8; ++r) {
    int pix = base + rbase + r;
    out[((size_t)b * 4096 + pix) * 16 + m] = fmaxf(acc[r] * sc + bi, 0.f);
  }
}

// pool1: 2x2 max on (320,64,64,16) + combine: ch<8: 0.5*(xm+xd) (path1 input),
// ch>=8: x_diff (path2 input). out (320, 32*32, 16)
__global__ void k_pool1(const float* __restrict__ in, float* __restrict__ out) {
  int i = blockIdx.x * blockDim.x + threadIdx.x;
  if (i >= 320 * 1024 * 16) return;
  int o = i & 15;
  int rest = i >> 4;
  int pix = rest & 1023, b = rest >> 10;
  int ph = pix >> 5, pw = pix & 31;
  size_t bb = (size_t)b * 4096;
  auto max4 = [&](int ch) {
    int h0 = ph * 2, w0 = pw * 2;
    float v0 = in[(bb + h0 * 64 + w0) * 16 + ch];
    float v1 = in[(bb + h0 * 64 + w0 + 1) * 16 + ch];
    float v2 = in[(bb + (h0 + 1) * 64 + w0) * 16 + ch];
    float v3 = in[(bb + (h0 + 1) * 64 + w0 + 1) * 16 + ch];
    return fmaxf(fmaxf(v0, v1), fmaxf(v2, v3));
  };
  out[i] = (o < 8) ? 0.5f * (max4(o) + max4(o + 8)) : max4(o);
}

// Stem conv2: fused conv21(path1)+conv22(x_diff), 7x7 stride1 pad3, BN+ReLU.
// in (320, 32*32, 16) NHWC, out (320, 32*32, 32).  K = 49 taps * 16 ch.
__global__ void k_conv2(const float* __restrict__ in, const float* __restrict__ Wp,
                        const float* __restrict__ sb, float* __restrict__ out) {
  extern __shared__ float lw[];  // 49*32*16
  for (int i = threadIdx.x; i < 49 * 32 * 16; i += blockDim.x) lw[i] = Wp[i];
  __syncthreads();
  int wid = threadIdx.x >> 5, lane = threadIdx.x & 31;
  int b = blockIdx.y;
  int tile = blockIdx.x * 4 + wid;
  int base = tile * 16;
  int m = lane & 15;
  int p = base + m, oh = p >> 5, ow = p & 31;
  int khalf = (lane < 16) ? 0 : 2;
  v8f acc0 = {}, acc1 = {};
  const float* bbase = in + (size_t)b * 1024 * 16;
#pragma unroll 1
  for (int kh = 0; kh < 7; ++kh) {
    int ih = oh + kh - 3;
#pragma unroll 1
    for (int kw = 0; kw < 7; ++kw) {
      int iw = ow + kw - 3;
      bool inb = (ih >= 0 && ih < 32 && iw >= 0 && iw < 32);
      const float* px = bbase + (size_t)(inb ? (ih * 32 + iw) : 0) * 16;
      float msk = inb ? 1.f : 0.f;
      const float* lwt = lw + ((kh * 7 + kw) * 32) * 16;
#pragma unroll
      for (int cb = 0; cb < 16; cb += 4) {
        int cbase = cb + khalf;
        v2f a = ldv2g(px + cbase);
        a[0] *= msk; a[1] *= msk;
        v2f b0 = *(const v2f*)&lwt[m * 16 + cbase];
        v2f b1 = *(const v2f*)&lwt[(16 + m) * 16 + cbase];
        acc0 = wmma4(a, b0, acc0);
        acc1 = wmma4(a, b1, acc1);
      }
    }
  }
  int rbase = (lane < 16) ? 0 : 8;
#pragma unroll
  for (int r = 0; r < 8; ++r) {
    int pix = base + rbase + r;
    size_t ob = ((size_t)b * 1024 + pix) * 32;
    out[ob + m]      = fmaxf(acc0[r] * sb[m] + sb[32 + m], 0.f);
    out[ob + 16 + m] = fmaxf(acc1[r] * sb[16 + m] + sb[48 + m], 0.f);
  }
}

// pool2 + 0.5*(path1+path2): in (320, 32*32, 32) -> xf (320, 16*16, 16)
__global__ void k_pool2(const float* __restrict__ in, float* __restrict__ out) {
  int i = blockIdx.x * blockDim.x + threadIdx.x;
  if (i >= 320 * 256 * 16) return;
  int o = i & 15;
  int rest = i >> 4;
  int pix = rest & 255, b = rest >> 8;
  int ph = pix >> 4, pw = pix & 15;
  size_t bb = (size_t)b * 1024;
  auto max4 = [&](int ch) {
    int h0 = ph * 2, w0 = pw * 2;
    float v0 = in[(bb + h0 * 32 + w0) * 32 + ch];
    float v1 = in[(bb + h0 * 32 + w0 + 1) * 32 + ch];
    float v2 = in[(bb + (h0 + 1) * 32 + w0) * 32 + ch];
    float v3 = in[(bb + (h0 + 1) * 32 + w0 + 1) * 32 + ch];
    return fmaxf(fmaxf(v0, v1), fmaxf(v2, v3));
  };
  out[i] = 0.5f * (max4(o) + max4(o + 16));
}

// conv3d: xf (2,160,16,16 | 16ch NHWC) -> (2,160,256,64), 3x5x5 pad(1,2,2),
// BN folded (no ReLU).  K = 75 taps * 16 ch.  Weights stream from L2.
__global__ void k_conv3(const float* __restrict__ xf, const float* __restrict__ Wp,
                        const float* __restrict__ sb, float* __restrict__ out) {
  int wid = threadIdx.x >> 5, lane = threadIdx.x & 31;
  int wg = blockIdx.x * 4 + wid;
  int tile = wg & 15;
  int d = (wg >> 4) % 160;
  int n = wg / 2560;
  int base = tile * 16;
  int m = lane & 15;
  int p = base + m, oh = p >> 4, ow = p & 15;
  int khalf = (lane < 16) ? 0 : 2;
  v8f a0 = {}, a1 = {}, a2 = {}, a3 = {};
  const float* wr = Wp;  // advances by 64*16 per tap
#pragma unroll 1
  for (int kd = 0; kd < 3; ++kd) {
    int id = d + kd - 1;
    bool dok = (id >= 0 && id < 160);
    const float* dbasep = xf + (size_t)(n * 160 + (dok ? id : 0)) * 256 * 16;
#pragma unroll 1
    for (int kh = 0; kh < 5; ++kh) {
      int ih = oh + kh - 2;
#pragma unroll 1
      for (int kw = 0; kw < 5; ++kw) {
        int iw = ow + kw - 2;
        bool inb = dok && (ih >= 0 && ih < 16 && iw >= 0 && iw < 16);
        const float* px = dbasep + (size_t)(inb ? (ih * 16 + iw) : 0) * 16;
        float msk = inb ? 1.f : 0.f;
        __builtin_prefetch(wr + 1024, 0, 0);  // next tap's weights
#pragma unroll
        for (int cb = 0; cb < 16; cb += 4) {
          int cbase = cb + khalf;
          v2f a = ldv2g(px + cbase);
          a[0] *= msk; a[1] *= msk;
          v2f b0 = ldv2g(wr + m * 16 + cbase);
          v2f b1 = ldv2g(wr + (16 + m) * 16 + cbase);
          v2f b2 = ldv2g(wr + (32 + m) * 16 + cbase);
          v2f b3 = ldv2g(wr + (48 + m) * 16 + cbase);
          a0 = wmma4(a, b0, a0);
          a1 = wmma4(a, b1, a1);
          a2 = wmma4(a, b2, a2);
          a3 = wmma4(a, b3, a3);
        }
        wr += 1024;
      }
    }
  }
  int rbase = (lane < 16) ? 0 : 8;
  size_t rowb = (size_t)(n * 160 + d) * 256;
#pragma unroll
  for (int r = 0; r < 8; ++r) {
    size_t ob = (rowb + base + rbase + r) * 64;
    out[ob + m]      = a0[r] * sb[m]      + sb[64 + m];
    out[ob + 16 + m] = a1[r] * sb[16 + m] + sb[80 + m];
    out[ob + 32 + m] = a2[r] * sb[32 + m] + sb[96 + m];
    out[ob + 48 + m] = a3[r] * sb[48 + m] + sb[112 + m];
  }
}

// mask-normalize + spatial mean + pos_emb:
// seq[n,l,c] = 0.5 * sum_p(xf*sig(xf)) / sum_p(sig(xf)) + pos[l,c]
__global__ void k_maskmean(const float* __restrict__ xf3d, const float* __restrict__ pos,
                           float* __restrict__ seq) {
  int bid = blockIdx.x;  // n*160+d
  int d = bid % 160;
  int t = threadIdx.x;   // channel (64)
  size_t rb = (size_t)bid * 256 * 64;
  float s1 = 0.f, s2 = 0.f;
  for (int p = 0; p < 256; ++p) {
    float v = xf3d[rb + (size_t)p * 64 + t];
    float sg = sigm_(v);
    s1 += sg;
    s2 += v * sg;
  }
  seq[(size_t)bid * 64 + t] = 0.5f * s2 / s1 + pos[d * 64 + t];
}

// ---------------------------------------------------------------------------
// Mamba stage kernels
// ---------------------------------------------------------------------------
__global__ void k_ln(const float* __restrict__ seq, const float* __restrict__ g,
                     const float* __restrict__ b, float* __restrict__ u) {
  __shared__ float red[64];
  __shared__ float mean_s, var_s;
  int row = blockIdx.x, t = threadIdx.x;
  float v = seq[(size_t)row * 64 + t];
  red[t] = v;
  __syncthreads();
  for (int off = 32; off > 0; off >>= 1) {
    if (t < off) red[t] += red[t + off];
    __syncthreads();
  }
  if (t == 0) mean_s = red[0] * (1.f / 64.f);
  __syncthreads();
  float dv = v - mean_s;
  red[t] = dv * dv;
  __syncthreads();
  for (int off = 32; off > 0; off >>= 1) {
    if (t < off) red[t] += red[t + off];
    __syncthreads();
  }
  if (t == 0) var_s = red[0] * (1.f / 64.f);
  __syncthreads();
  u[(size_t)row * 64 + t] = dv * rsqrtf(var_s + EPS_) * g[t] + b[t];
}

// Generic f32 WMMA GEMM: C[M,N] = A[M,K] * W[N,K]^T (+ Cadd).  M%16==0, K%4==0.
__global__ void k_gemm(const float* __restrict__ A, const float* __restrict__ W,
                       const float* __restrict__ Cadd, float* __restrict__ C,
                       int M, int N, int K) {
  int wave = (blockIdx.x * blockDim.x + threadIdx.x) >> 5;
  int lane = threadIdx.x & 31;
  int ntiles = (N + 15) >> 4;
  int mt = wave / ntiles, nt = wave - mt * ntiles;
  if (mt * 16 >= M) return;
  int mrow = mt * 16 + (lane & 15);
  int ncol = nt * 16 + (lane & 15);
  int khalf = (lane < 16) ? 0 : 2;
  const float* Arow = A + (size_t)mrow * K;
  bool nok = (ncol < N);
  const float* Wrow = W + (size_t)(nok ? ncol : 0) * K;
  __builtin_prefetch(Wrow, 0, 0);
  v8f acc = {};
#pragma unroll 1
  for (int k = 0; k < K; k += 4) {
    v2f a = ldv2g(Arow + k + khalf);
    v2f b;
    if (nok) b = ldv2g(Wrow + k + khalf);
    else { b[0] = 0.f; b[1] = 0.f; }
    acc = wmma4(a, b, acc);
  }
  if (nok) {
    int rbase = mt * 16 + ((lane < 16) ? 0 : 8);
#pragma unroll
    for (int r = 0; r < 8; ++r) {
      int cm = rbase + r;
      float v = acc[r];
      if (Cadd) v += Cadd[(size_t)cm * N + ncol];
      C[(size_t)cm * N + ncol] = v;
    }
  }
}

// causal depthwise conv1d (K=4) + SiLU, with optional sequence flip (bwd dir).
// xi = xz[...,0:128] (row stride 256).  xs stored in scan-time order.
__global__ void k_conv1d(const float* __restrict__ xz, const float* __restrict__ cw,
                         const float* __restrict__ cb, float* __restrict__ xs,
                         int dirflag) {
  int i = blockIdx.x * blockDim.x + threadIdx.x;
  if (i >= 2 * 160 * 128) return;
  int dch = i & 127;
  int rest = i >> 7;
  int l = rest % 160, n = rest / 160;
  float s = cb[dch];
#pragma unroll
  for (int j = 0; j < 4; ++j) {
    int t = l - 3 + j;
    if (t >= 0) {
      int src = dirflag ? (159 - t) : t;
      s += cw[dch * 4 + j] * xz[(size_t)(n * 160 + src) * 256 + dch];
    }
  }
  xs[i] = s * sigm_(s);
}

// dt = softplus(dtr @ dtproj_w^T + dtproj_b); dtr = dbl[...,0:4]
__global__ void k_dt(const float* __restrict__ dbl, const float* __restrict__ dtw,
                     const float* __restrict__ dtb, float* __restrict__ dt) {
  int i = blockIdx.x * blockDim.x + threadIdx.x;
  if (i >= 320 * 128) return;
  int dch = i & 127, row = i >> 7;
  float s = dtb[dch];
#pragma unroll
  for (int j = 0; j < 4; ++j) s += dbl[(size_t)row * 100 + j] * dtw[dch * 4 + j];
  dt[i] = (s > 20.f) ? s : log1pf(__expf(s));
}

// selective scan: block per (n,d); lanes hold 48 states; per-step LDS reduce.
// Bm = dbl[...,4:52], Cm = dbl[...,52:100].  Output at flipped pos for bwd.
__global__ void k_scan(const float* __restrict__ xs, const float* __restrict__ dbl,
                       const float* __restrict__ dt, const float* __restrict__ A,
                       const float* __restrict__ Dp, float* __restrict__ yacc,
                       int dirflag, int accumulate) {
  __shared__ float red[64];
  int t = threadIdx.x;
  int d = blockIdx.x & 127, n = blockIdx.x >> 7;
  float h = 0.f;
  float As = (t < 48) ? A[d * 48 + t] : 0.f;
  float Dv = Dp[d];
  for (int l = 0; l < 160; ++l) {
    int row = n * 160 + l;
    float dtv = dt[(size_t)row * 128 + d];
    float xsv = xs[(size_t)row * 128 + d];
    float contrib = 0.f;
    if (t < 48) {
      float Bm = dbl[(size_t)row * 100 + 4 + t];
      float Cm = dbl[(size_t)row * 100 + 52 + t];
      h = __expf(dtv * As) * h + dtv * xsv * Bm;
      contrib = h * Cm;
    }
    red[t] = contrib;
    __syncthreads();
    for (int off = 32; off > 0; off >>= 1) {
      if (t < off) red[t] += red[t + off];
      __syncthreads();
    }
    if (t == 0) {
      int lo = dirflag ? (159 - l) : l;
      size_t oi = (size_t)(n * 160 + lo) * 128 + d;
      float y = red[0] + xsv * Dv;
      yacc[oi] = accumulate ? (yacc[oi] + y) : y;
    }
    __syncthreads();
  }
}

// (yf+yb) * silu(z), in place on yacc.  z = xz[...,128:256]
__global__ void k_combine(float* __restrict__ yacc, const float* __restrict__ xz) {
  int i = blockIdx.x * blockDim.x + threadIdx.x;
  if (i >= 320 * 128) return;
  int dch = i & 127, row = i >> 7;
  float z = xz[(size_t)row * 256 + 128 + dch];
  yacc[i] *= z * sigm_(z);
}

// tok -> codebook soft-quantize -> proj; proj_sum accumulate; seq += proj_sum
__global__ void k_head(float* __restrict__ seq, float* __restrict__ psum,
                       const float* __restrict__ hw, const float* __restrict__ hb,
                       const float* __restrict__ cb, int ncb,
                       const float* __restrict__ pw, const float* __restrict__ pb,
                       int first) {
  __shared__ float red[64];
  __shared__ float recon_s;
  int row = blockIdx.x, t = threadIdx.x;
  red[t] = seq[(size_t)row * 64 + t] * hw[t];
  __syncthreads();
  for (int off = 32; off > 0; off >>= 1) {
    if (t < off) red[t] += red[t + off];
    __syncthreads();
  }
  if (t == 0) {
    float tok = red[0] + hb[0];
    float mx = -1e30f;
    for (int i = 0; i < ncb; ++i) mx = fmaxf(mx, -fabsf(tok - cb[i]));
    float ssum = 0.f, rec = 0.f;
    for (int i = 0; i < ncb; ++i) {
      float e = __expf(-fabsf(tok - cb[i]) - mx);
      ssum += e;
      rec += e * cb[i];
    }
    recon_s = rec / ssum;
  }
  __syncthreads();
  float pr = recon_s * pw[t] + pb[t];
  float ps = first ? pr : (psum[(size_t)row * 64 + t] + pr);
  psum[(size_t)row * 64 + t] = ps;
  seq[(size_t)row * 64 + t] += ps;
}

// ---------------------------------------------------------------------------
// Host
// ---------------------------------------------------------------------------
extern "C" void kernel_launch(void* const* d_in, const int* in_sizes, int n_in,
                              void* d_out, int out_size, void* d_ws, size_t ws_size,
                              hipStream_t stream) {
  (void)out_size; (void)ws_size;
  // Locate x by its unique size (handles either flattening order of the dict).
  const long long XN = 2LL * 3 * 160 * 128 * 128;
  int pb = 1;
  const float* x = (const float*)d_in[0];
  if (!(n_in > 0 && (long long)in_sizes[0] == XN)) {
    x = (const float*)d_in[n_in - 1];
    pb = 0;
  }
  auto P = [&](int i) -> const float* { return (const float*)d_in[pb + i]; };
  // params leaves assumed in jax-pytree (sorted-key) order:
  //  [0..3] codebooks(2,4,8,16)  [4] pos_emb
  //  per-stage s (base 5+22s): bwd{A_log,D,conv_b,conv_w,dtproj_b,dtproj_w,xproj_w},
  //    fwd{same 7}, head_b, head_w, in_w, ln_b, ln_g, out_w, proj_b, proj_w
  //  stem (base 93): b11,b12,b21,b22,b3, bn11{b,g,m,v}, bn12{..}, bn21{..},
  //    bn22{..}, bn3{..}, w11,w12,w21,w22,w3
  auto SBase = [](int s) { return 5 + 22 * s; };
  const int ST = 93;

  float* ws = (float*)d_ws;
  size_t off = 0;
  auto alloc = [&](size_t nf) { float* r = ws + off; off += (nf + 63) & ~(size_t)63; return r; };
  float* convA = alloc(320ull * 4096 * 16);  // conv1 out; reused as conv2 out
  float* pool1 = alloc(320ull * 1024 * 16);
  float* xf    = alloc(320ull * 256 * 16);
  float* xf3d  = alloc(2ull * 160 * 256 * 64);
  float* pw1   = alloc(49 * 16 * 16);
  float* sb1   = alloc(32);
  float* pw2   = alloc(49 * 32 * 16);
  float* sb2   = alloc(64);
  float* pw3   = alloc(75 * 64 * 16);
  float* sb3   = alloc(128);
  float* Aneg  = alloc(8 * 128 * 48);
  float* u     = alloc(320 * 64);
  float* xz    = alloc(320 * 256);
  float* xs    = alloc(320 * 128);
  float* dbl   = alloc(320 * 100);
  float* dtb   = alloc(320 * 128);
  float* yacc  = alloc(320 * 128);
  float* psum  = alloc(320 * 64);
  float* seq   = (float*)d_out;  // seq lives in d_out (fully rewritten each call)

  auto gemm = [&](const float* A, const float* W, const float* Cadd, float* C,
                  int M, int N, int K) {
    int waves = ((M + 15) / 16) * ((N + 15) / 16);
    k_gemm<<<(waves + 3) / 4, 128, 0, stream>>>(A, W, Cadd, C, M, N, K);
  };

  // ---- prep ----
  k_pack_w1<<<(49 * 16 * 16 + 255) / 256, 256, 0, stream>>>(P(ST + 25), P(ST + 26), pw1);
  k_pack_w2<<<(49 * 32 * 16 + 255) / 256, 256, 0, stream>>>(P(ST + 27), P(ST + 28), pw2);
  k_pack_w3<<<(75 * 64 * 16 + 255) / 256, 256, 0, stream>>>(P(ST + 29), pw3);
  // bn dicts flatten as {b,g,m,v}; bn11@ST+5, bn12@+9, bn21@+13, bn22@+17, bn3@+21
  k_make_sb<<<1, 32, 0, stream>>>(P(ST + 6),  P(ST + 5),  P(ST + 7),  P(ST + 8),  P(ST + 0), sb1,      sb1 + 16, 8);
  k_make_sb<<<1, 32, 0, stream>>>(P(ST + 10), P(ST + 9),  P(ST + 11), P(ST + 12), P(ST + 1), sb1 + 8,  sb1 + 24, 8);
  k_make_sb<<<1, 32, 0, stream>>>(P(ST + 14), P(ST + 13), P(ST + 15), P(ST + 16), P(ST + 2), sb2,      sb2 + 32, 16);
  k_make_sb<<<1, 32, 0, stream>>>(P(ST + 18), P(ST + 17), P(ST + 19), P(ST + 20), P(ST + 3), sb2 + 16, sb2 + 48, 16);
  k_make_sb<<<1, 64, 0, stream>>>(P(ST + 22), P(ST + 21), P(ST + 23), P(ST + 24), P(ST + 4), sb3,      sb3 + 64, 64);
  for (int s = 0; s < 4; ++s)
    for (int dir = 0; dir < 2; ++dir)
      k_negexp<<<24, 256, 0, stream>>>(P(SBase(s) + (dir ? 0 : 7)),
                                       Aneg + (s * 2 + dir) * 6144, 6144);

  // ---- stem ----
  k_conv1<<<dim3(64, 320), 128, 49 * 16 * 16 * 4, stream>>>(x, pw1, sb1, convA);
  k_pool1<<<(320 * 1024 * 16 + 255) / 256, 256, 0, stream>>>(convA, pool1);
  k_conv2<<<dim3(16, 320), 128, 49 * 32 * 16 * 4, stream>>>(pool1, pw2, sb2, convA);
  k_pool2<<<(320 * 256 * 16 + 255) / 256, 256, 0, stream>>>(convA, xf);
  k_conv3<<<1280, 128, 0, stream>>>(xf, pw3, sb3, xf3d);
  k_maskmean<<<320, 64, 0, stream>>>(xf3d, P(4), seq);

  // ---- stages ----
  const int ncbs[4] = {2, 4, 8, 16};
  for (int s = 0; s < 4; ++s) {
    int B0 = SBase(s);
    k_ln<<<320, 64, 0, stream>>>(seq, P(B0 + 18), P(B0 + 17), u);
    gemm(u, P(B0 + 16), nullptr, xz, 320, 256, 64);  // in_w
    for (int dir = 0; dir < 2; ++dir) {
      int DB = B0 + (dir ? 0 : 7);  // fwd leaves at +7.., bwd at +0..
      k_conv1d<<<(2 * 160 * 128 + 255) / 256, 256, 0, stream>>>(xz, P(DB + 3), P(DB + 2), xs, dir);
      gemm(xs, P(DB + 6), nullptr, dbl, 320, 100, 128);  // xproj_w
      k_dt<<<(320 * 128 + 255) / 256, 256, 0, stream>>>(dbl, P(DB + 5), P(DB + 4), dtb);
      k_scan<<<256, 64, 0, stream>>>(xs, dbl, dtb, Aneg + (s * 2 + dir) * 6144,
                                     P(DB + 1), yacc, dir, dir);
    }
    k_combine<<<(320 * 128 + 255) / 256, 256, 0, stream>>>(yacc, xz);
    gemm(yacc, P(B0 + 19), seq, seq, 320, 64, 128);  // out_w + residual
    k_head<<<320, 64, 0, stream>>>(seq, psum, P(B0 + 15), P(B0 + 14), P(s), ncbs[s],
                                   P(B0 + 21), P(B0 + 20), s == 0 ? 1 : 0);
  }
}